// SCON_57200374448428
// MI455X (gfx1250) — compile-verified
//
#include <hip/hip_runtime.h>
#include <math.h>

typedef float v2f __attribute__((ext_vector_type(2)));
typedef float v4f __attribute__((ext_vector_type(4)));
typedef float v8f __attribute__((ext_vector_type(8)));
typedef unsigned int u32x4 __attribute__((ext_vector_type(4)));
typedef int i32x4 __attribute__((ext_vector_type(4)));
typedef int i32x8 __attribute__((ext_vector_type(8)));

namespace {
constexpr int   NSTEP  = 10001;   // N time points (t = 0..10000)
constexpr int   NCHUNK = 200;     // 10000 / 50
constexpr int   CHUNK  = 50;      // obs interval
constexpr int   NOBS   = 201;
constexpr float DTC    = 0.1f;
constexpr float GASR   = 0.008314f;

// workspace layout (floats)
constexpr int OFF_CO  = 0;                           // 12 * NSTEP       = 120012
constexpr int OFF_C1  = 12 * NSTEP;                  // 4 * NOBS         = 804
constexpr int OFF_RS  = OFF_C1 + 4 * NOBS;           // NCHUNK*CHUNK*16  = 160000
constexpr int OFF_PB  = OFF_RS + NCHUNK * CHUNK * 16;// NCHUNK*16        = 3200
constexpr int OFF_ACC = OFF_PB + NCHUNK * 16;        // NCHUNK*32*64     = 409600
} // namespace

// ---------------------------------------------------------------------------
// TDM: issue a 2-D tensor_load_to_lds per the CDNA5 D# spec (data_size = 4B,
// count=1, type=2). 1-D copies use tensor_d1=1, tile_d1=0.
// This toolchain (clang-23/therock-10.0) uses the 6-arg builtin:
//   (u32x4 g0, i32x8 g1, i32x4 g2, i32x4 g3, i32x8 extra, i32 cpol)
// ---------------------------------------------------------------------------
__device__ __forceinline__ void tdm_load_2d(const void* gaddr, unsigned lds_byte,
                                            unsigned tensor_d0, unsigned tensor_d1,
                                            unsigned tile_d0, unsigned tile_d1,
                                            unsigned long long stride0_elems) {
  unsigned long long ga = (unsigned long long)gaddr;
  u32x4 g0;
  g0.x = 1u;                                            // count=1, user mode
  g0.y = lds_byte;                                      // LDS byte address
  g0.z = (unsigned)(ga & 0xFFFFFFFFu);                  // global_addr[31:0]
  g0.w = (unsigned)((ga >> 32) & 0x01FFFFFFu)           // global_addr[56:32]
       | (2u << 30);                                    // type = 2 ("image")
  i32x8 g1;
  g1[0] = (int)(2u << 16);                              // data_size = 4 bytes
  g1[1] = (int)((tensor_d0 & 0xFFFFu) << 16);           // tensor_dim0[15:0]
  g1[2] = (int)((tensor_d0 >> 16) & 0xFFFFu)
        | (int)((tensor_d1 & 0xFFFFu) << 16);           // dim0 hi | dim1 lo
  g1[3] = (int)((tensor_d1 >> 16) & 0xFFFFu)
        | (int)(tile_d0 << 16);                         // dim1 hi | tile_dim0
  g1[4] = (int)(tile_d1 & 0xFFFFu);                     // tile_dim1 (tile_dim2=0)
  g1[5] = (int)(stride0_elems & 0xFFFFFFFFu);           // tensor_dim0_stride lo
  g1[6] = (int)((stride0_elems >> 32) & 0xFFFFu);       // stride hi (dim1_stride=0)
  g1[7] = 0;
  i32x4 z4 = {0, 0, 0, 0};
  i32x8 z8 = {0, 0, 0, 0, 0, 0, 0, 0};
  __builtin_amdgcn_tensor_load_to_lds(g0, g1, z4, z4, z8, 0);
}

// Opaque zero: the compiler cannot prove its value, so a guarded store through
// it makes a TDM-written __shared__ array "possibly stored" (blocks the
// loads-of-never-stored-LDS => undef folding) without any runtime store.
__device__ __forceinline__ unsigned opaque_zero() {
  unsigned g = 0;
  asm volatile("" : "+s"(g));
  return g;
}

// ---------------------------------------------------------------------------
// K1: per-step coefficients  M_t = I + DT*W(t)  (row-major, pad 4, DT*b in col 3)
// and observation row C1[n] at every 50th step.
// ---------------------------------------------------------------------------
__global__ void coef_kernel(const float* __restrict__ th, float* __restrict__ ws) {
  int t = blockIdx.x * blockDim.x + threadIdx.x;
  if (t >= NSTEP) return;
  float times = 0.1f * (float)t;
  const float W1 = (float)(6.283185307179586 / 24.0);
  const float W2 = (float)(6.283185307179586 / (24.0 * 365.0));
  float sy   = __sinf(W2 * times);
  float temp = 283.0f + 5.0f * times * (1.0f / (80.0f * 24.0f * 365.0f))
             + 10.0f * __sinf(W1 * times) + 10.0f * sy;
  float I_S  = 0.001f  + 0.0005f  * sy;
  float I_D  = 0.0001f + 0.00005f * sy;
  float dinv = 1.0f / temp - (1.0f / 283.0f);
  float kS = th[0] * __expf(-th[3] / GASR * dinv);
  float kD = th[1] * __expf(-th[4] / GASR * dinv);
  float kM = th[2] * __expf(-th[5] / GASR * dinv);
  float aSD = th[6], aDS = th[7], aM = th[8], aMSC = th[9], uM = th[10];
  float* c = ws + OFF_CO + t * 12;
  c[0] = 1.0f - DTC * kS;       c[1] = DTC * aDS * kD;         c[2]  = DTC * aM * aMSC * kM;          c[3]  = DTC * I_S;
  c[4] = DTC * aSD * kS;        c[5] = 1.0f - DTC * (uM + kD); c[6]  = DTC * aM * (1.0f - aMSC) * kM; c[7]  = DTC * I_D;
  c[8] = 0.0f;                  c[9] = DTC * uM;               c[10] = 1.0f - DTC * kM;               c[11] = 0.0f;
  if (t % CHUNK == 0) {
    float* o = ws + OFF_C1 + (t / CHUNK) * 4;
    o[0] = (1.0f - aSD) * kS;
    o[1] = (1.0f - aDS) * kD;
    o[2] = (1.0f - aM)  * kM;
    o[3] = 0.0f;
  }
}

// ---------------------------------------------------------------------------
// K2: per-chunk prefix structure (serial 3x3 products, 200 independent chunks).
//   Rs_tau[j][i] = R_tau[i][j]*s[j]  (WMMA B-layout, pad 4)
//   accb  = sum_tau R_tau * (DT*b_tau);   P_n = full-chunk product (A-layout)
// ---------------------------------------------------------------------------
__global__ void chunk_kernel(const float* __restrict__ th, float* __restrict__ ws) {
  int n = blockIdx.x * blockDim.x + threadIdx.x;
  if (n >= NCHUNK) return;
  float sv[3];
  sv[0] = sqrtf(fmaxf(th[11], 1e-6f) * DTC);
  sv[1] = sqrtf(fmaxf(th[12], 1e-6f) * DTC);
  sv[2] = sqrtf(fmaxf(th[13], 1e-6f) * DTC);
  float R[9]    = {1.f,0.f,0.f, 0.f,1.f,0.f, 0.f,0.f,1.f};
  float accb[3] = {0.f,0.f,0.f};
  for (int tau = CHUNK; tau >= 1; --tau) {
    int t = n * CHUNK + tau;
    const float* M = ws + OFF_CO + t * 12;
    float* rs = ws + OFF_RS + (n * CHUNK + tau - 1) * 16;
    #pragma unroll
    for (int j = 0; j < 3; ++j) {
      #pragma unroll
      for (int i = 0; i < 3; ++i) rs[j * 4 + i] = R[i * 3 + j] * sv[j];
      rs[j * 4 + 3] = 0.0f;
    }
    #pragma unroll
    for (int i = 0; i < 3; ++i)
      accb[i] += R[i * 3 + 0] * M[3] + R[i * 3 + 1] * M[7];
    float Rn[9];
    #pragma unroll
    for (int i = 0; i < 3; ++i)
      #pragma unroll
      for (int j = 0; j < 3; ++j)
        Rn[i * 3 + j] = R[i*3+0]*M[0*4+j] + R[i*3+1]*M[1*4+j] + R[i*3+2]*M[2*4+j];
    #pragma unroll
    for (int q = 0; q < 9; ++q) R[q] = Rn[q];
  }
  float* pb = ws + OFF_PB + n * 16;
  #pragma unroll
  for (int i = 0; i < 3; ++i) {
    pb[i*4+0] = R[i*3+0]; pb[i*4+1] = R[i*3+1]; pb[i*4+2] = R[i*3+2]; pb[i*4+3] = 0.0f;
    pb[12 + i] = accb[i];
  }
  pb[15] = 0.0f;
}

// ---------------------------------------------------------------------------
// K3: per (chunk, 16-batch tile) wave: TDM-stage the chunk's Rs table into LDS,
// then 50 chained V_WMMA_F32_16X16X4_F32 accumulations over the noise stream.
// Plain (non-volatile) LDS reads so they lower to ds_load; the opaque guarded
// store keeps them from being folded to undef.
// ---------------------------------------------------------------------------
__global__ void __launch_bounds__(32) gemm_kernel(const float* __restrict__ noise,
                                                  const float* __restrict__ ws,
                                                  float* __restrict__ wsacc) {
  __shared__ float lds_rs[CHUNK * 16];                 // 3200 B
  const int  blk  = blockIdx.x;
  const int  n    = blk >> 5;      // chunk
  const int  tile = blk & 31;      // batch tile
  const int  l    = threadIdx.x;
  const bool lo   = l < 16;
  const int  m    = l & 15;
  const int  b    = tile * 16 + m;
  const int  ic   = (m < 3) ? m : 0;
  const float imask = (m < 3) ? 1.0f : 0.0f;
  const int  ib0  = (lo ? 0 : 8) + ic;                 // Rs row j=0 | j=2
  const int  ib1  = 4 + ic;                            // Rs row j=1

  // Async DMA: Rs table for this chunk (1-D, 800 dwords) -> LDS.
  tdm_load_2d(ws + OFF_RS + n * CHUNK * 16, 0u,
              CHUNK * 16, 1u, CHUNK * 16, 0u, CHUNK * 16);

  // Never executes; makes lds_rs "possibly stored" for alias analysis.
  unsigned gg = opaque_zero();
  if (gg) lds_rs[gg & 511u] = 0.0f;

  const float* np = noise + (long)b * (3L * (NSTEP - 1)) + (long)(n * CHUNK) * 3;

  __builtin_amdgcn_s_wait_tensorcnt(0);
  asm volatile("" ::: "memory");

  v8f acc = {0.f,0.f,0.f,0.f,0.f,0.f,0.f,0.f};
  #pragma unroll 5
  for (int tau = 0; tau < CHUNK; ++tau) {
    const float* p = np + tau * 3;
    float e0 = p[0], e1 = p[1], e2 = p[2];             // contiguous b96/b128 loads
    v2f A;
    A.x = lo ? e0 : e2;
    A.y = lo ? e1 : 0.0f;
    float b0 = lds_rs[tau * 16 + ib0];                 // ds_load_b32
    float b1 = lds_rs[tau * 16 + ib1];
    v2f B;
    B.x = b0 * imask;
    B.y = lo ? (b1 * imask) : 0.0f;
    acc = __builtin_amdgcn_wmma_f32_16x16x4_f32(false, A, false, B,
                                                (short)0, acc, false, false);
  }
  // acc layout: vgpr v, lanes0-15 -> (row v, col=lane); lanes16-31 -> (row v+8).
  if (m < 3) {
    float* o = wsacc + (long)(n * 32 + tile) * 64;     // [batch row 0..15][dim pad4]
    const int r0 = lo ? 0 : 8;
    #pragma unroll
    for (int v = 0; v < 8; ++v) o[(r0 + v) * 4 + m] = acc[v];
  }
}

// ---------------------------------------------------------------------------
// K4: residual 200-step scan per 16-batch tile, one wave each.
//   x_{n+1} = P_n x_n + (accb_n + gemm_acc_n[b]),  obs emitted each step.
// ACC block (2-D strided: 200 rows x 256B, 8KB stride) and P/accb table are
// TDM-staged into LDS; plain ds reads + opaque guard; state feedback via
// ds_bpermute.
// ---------------------------------------------------------------------------
__global__ void __launch_bounds__(32) scan_kernel(const float* __restrict__ x0,
                                                  const float* __restrict__ ws,
                                                  const float* __restrict__ wsacc,
                                                  float* __restrict__ out) {
  __shared__ float lds_buf[NCHUNK * 64 + NCHUNK * 16]; // 51200B ACC + 12800B PB
  constexpr unsigned PB_F = NCHUNK * 64;               // float offset of PB region
  const int  tile = blockIdx.x;
  const int  l    = threadIdx.x;
  const bool lo   = l < 16;
  const int  m    = l & 15;
  const int  b    = tile * 16 + m;
  const int  mc   = (m < 3) ? m : 0;
  const float mmask = (m < 3) ? 1.0f : 0.0f;

  // 2-D DMA: this tile's ACC rows (64 dwords each, stride 2048 dwords).
  tdm_load_2d(wsacc + tile * 64, 0u, 64u, NCHUNK, 64u, NCHUNK, 2048u);
  // 1-D DMA: P/accb table (3200 dwords).
  tdm_load_2d(ws + OFF_PB, PB_F * 4u, NCHUNK * 16, 1u, NCHUNK * 16, 0u, NCHUNK * 16);

  // Never executes; makes lds_buf "possibly stored" for alias analysis.
  unsigned gg = opaque_zero();
  if (gg) lds_buf[gg & 4095u] = 0.0f;

  const float xa = x0[b*3+0], xb = x0[b*3+1], xc = x0[b*3+2];
  v2f B;
  B.x = lo ? xa : xc;
  B.y = lo ? xb : 0.0f;
  {
    const float* c1 = ws + OFF_C1;
    if (lo) {
      float y3 = c1[0]*xa + c1[1]*xb + c1[2]*xc;
      v4f o = {xa, xb, xc, y3};
      *(v4f*)(out + (long)b * (NOBS * 4)) = o;
    }
  }

  __builtin_amdgcn_s_wait_tensorcnt(0);
  asm volatile("" ::: "memory");

  const int offA = lo ? 0 : 2;
  for (int n = 0; n < NCHUNK; ++n) {
    const float* pb = lds_buf + PB_F + n * 16;
    float a0 = pb[mc*4 + offA];
    float a1 = pb[mc*4 + 1];
    v2f A;
    A.x = a0 * mmask;
    A.y = lo ? (a1 * mmask) : 0.0f;
    const float* ac = lds_buf + n * 64 + m * 4;        // [batch m][dim pad4]
    float c0  = pb[12] + ac[0];
    float c1v = pb[13] + ac[1];
    float c2  = pb[14] + ac[2];
    v8f C = { lo ? c0 : 0.0f, lo ? c1v : 0.0f, lo ? c2 : 0.0f,
              0.0f, 0.0f, 0.0f, 0.0f, 0.0f };
    v8f D = __builtin_amdgcn_wmma_f32_16x16x4_f32(false, A, false, B,
                                                  (short)0, C, false, false);
    const float* c1 = ws + OFF_C1 + (n + 1) * 4;
    if (lo) {
      float y3 = c1[0]*D[0] + c1[1]*D[1] + c1[2]*D[2];
      v4f o = {D[0], D[1], D[2], y3};
      *(v4f*)(out + (long)b * (NOBS * 4) + (n + 1) * 4) = o;
    }
    // feedback: B row0 = D row0 | D row2 (shifted), B row1 = D row1 | 0
    float t2 = __int_as_float(__builtin_amdgcn_ds_bpermute(m << 2, __float_as_int(D[2])));
    B.x = lo ? D[0] : t2;
    B.y = lo ? D[1] : 0.0f;
  }
}

extern "C" void kernel_launch(void* const* d_in, const int* in_sizes, int n_in,
                              void* d_out, int out_size, void* d_ws, size_t ws_size,
                              hipStream_t stream) {
  const float* theta = (const float*)d_in[0];
  const float* x0    = (const float*)d_in[1];
  const float* noise = (const float*)d_in[2];
  float* ws  = (float*)d_ws;     // needs ~2.8 MB
  float* out = (float*)d_out;
  coef_kernel <<<(NSTEP + 255) / 256, 256, 0, stream>>>(theta, ws);
  chunk_kernel<<<1, 256, 0, stream>>>(theta, ws);
  gemm_kernel <<<NCHUNK * 32, 32, 0, stream>>>(noise, ws, ws + OFF_ACC);
  scan_kernel <<<32, 32, 0, stream>>>(x0, ws, ws + OFF_ACC, out);
}